// PennyLaneQuantumClassifier_3788161155351
// MI455X (gfx1250) — compile-verified
//
#include <hip/hip_runtime.h>

#define NQ 10
#define NL 4
#define DIM 1024
#define BATCH 32768

typedef __attribute__((ext_vector_type(2))) float v2f;
typedef __attribute__((ext_vector_type(8))) float v8f;

struct c32 { float x, y; };
__device__ inline c32 cmul(c32 a, c32 b){ return { a.x*b.x - a.y*b.y, a.x*b.y + a.y*b.x }; }
__device__ inline c32 cadd(c32 a, c32 b){ return { a.x + b.x, a.y + b.y }; }

// ---------------------------------------------------------------------------
// Kernel 1: simulate the circuit on basis state |j>, j = blockIdx.x in [0,10).
// Statevector (1024 complex) lives in LDS. Writes column j of W (1024 x 10).
// Rolled loops on purpose: this is a tiny one-off precompute; keep code small.
// ---------------------------------------------------------------------------
__global__ __launch_bounds__(512) void qc_sim_basis(const float* __restrict__ theta,
                                                    float2* __restrict__ W) {
  __shared__ float2 amp[DIM];
  const int j = blockIdx.x;   // basis index 0..9
  const int t = threadIdx.x;  // 0..511

  amp[t]       = (t == j) ? make_float2(1.f, 0.f) : make_float2(0.f, 0.f);
  amp[t + 512] = make_float2(0.f, 0.f);
  __syncthreads();

#pragma unroll 1
  for (int l = 0; l < NL; ++l) {
    // single-qubit Euler rotations RZ(a)RX(b)RZ(c) on every wire
#pragma unroll 1
    for (int q = 0; q < NQ; ++q) {
      const float* th = theta + (l * NQ + q) * 3;
      float a = th[0], b = th[1], c = th[2];
      float cb, sb, cp, sp, cm, sm;
      sincosf(0.5f * b, &sb, &cb);
      sincosf(0.5f * (a + c), &sp, &cp);
      sincosf(0.5f * (a - c), &sm, &cm);
      c32 u00 = {  cp * cb, -sp * cb };   //  e^{-i(a+c)/2} cos(b/2)
      c32 u01 = {  sm * sb, -cm * sb };   // -i e^{ i(a-c)/2} sin(b/2)
      c32 u10 = { -sm * sb, -cm * sb };   // -i e^{-i(a-c)/2} sin(b/2)
      c32 u11 = {  cp * cb,  sp * cb };   //  e^{ i(a+c)/2} cos(b/2)

      const int sh = 9 - q;               // wire q = bit (9-q)
      const int s  = 1 << sh;
      const int i0 = ((t >> sh) << (sh + 1)) | (t & (s - 1));
      const int i1 = i0 + s;
      float2 f0 = amp[i0], f1 = amp[i1];
      c32 a0 = { f0.x, f0.y }, a1 = { f1.x, f1.y };
      c32 r0 = cadd(cmul(u00, a0), cmul(u01, a1));
      c32 r1 = cadd(cmul(u10, a0), cmul(u11, a1));
      amp[i0] = make_float2(r0.x, r0.y);
      amp[i1] = make_float2(r1.x, r1.y);
      __syncthreads();
    }
    // CNOT ladder: control wire q (bit 9-q), target wire q+1 (bit 8-q)
#pragma unroll 1
    for (int q = 0; q < NQ - 1; ++q) {
      const int pt = 8 - q, pc = 9 - q;
      if (t < 256) {
        int base = ((t >> pt) << (pt + 2)) | (t & ((1 << pt) - 1));
        int i0 = base | (1 << pc);          // control = 1, target = 0
        int i1 = i0 | (1 << pt);            // control = 1, target = 1
        float2 tmp = amp[i0];
        amp[i0] = amp[i1];
        amp[i1] = tmp;
      }
      __syncthreads();
    }
  }

  W[t * NQ + j]         = amp[t];
  W[(t + 512) * NQ + j] = amp[t + 512];
}

// ---------------------------------------------------------------------------
// Kernel 2: ReA[i][j] = sum_k sign(k) * Re( conj(W[k][i]) * W[k][j] ),
// sign(k) = +1 if MSB(k)==0 else -1. Padded to 16x16 with zeros.
// ---------------------------------------------------------------------------
__global__ __launch_bounds__(256) void qc_build_A(const float2* __restrict__ W,
                                                  float* __restrict__ ReA) {
  const int t = threadIdx.x;        // 0..255
  const int i = t >> 4, j = t & 15;
  float acc = 0.f;
  if (i < NQ && j < NQ) {
#pragma unroll 4
    for (int k = 0; k < DIM; ++k) {
      float sgn = (k & 512) ? -1.f : 1.f;
      float2 wi = W[k * NQ + i];
      float2 wj = W[k * NQ + j];
      acc += sgn * (wi.x * wj.x + wi.y * wj.y);
    }
  }
  ReA[i * 16 + j] = acc;
}

// ---------------------------------------------------------------------------
// Kernel 3 (hot): 16 batch elements per wave32.
//   v = triple-normalized tanh(x*scale) (collapsed to one scale factor),
//   y = ReA * v via 3 chained V_WMMA_F32_16X16X4_F32 (K chunk 12..15 is all
//   zero and elided), z = v . y via FMAs + one cross-half shuffle add.
// ---------------------------------------------------------------------------
__global__ __launch_bounds__(256) void qc_batch(const float* __restrict__ x,
                                                const float* __restrict__ scale,
                                                const float* __restrict__ out_w,
                                                const float* __restrict__ out_b,
                                                const float* __restrict__ ReA,
                                                float* __restrict__ out) {
  const int lane = threadIdx.x & 31;
  const int wave = (blockIdx.x * blockDim.x + threadIdx.x) >> 5;
  const int col  = lane & 15;            // batch element within wave tile
  const bool hi  = lane >= 16;
  const int e    = wave * 16 + col;
  const float* xe = x + (size_t)e * NQ;

  // ---- v = tanh(x*scale) normalized; the reference's three normalizations
  // compose into a single scalar factor: 1/sqrt((s+eps)*(s2+eps)*s3).
  float v0 = tanhf(xe[0] * scale[0]);
  float v1 = tanhf(xe[1] * scale[1]);
  float v2 = tanhf(xe[2] * scale[2]);
  float v3 = tanhf(xe[3] * scale[3]);
  float v4 = tanhf(xe[4] * scale[4]);
  float v5 = tanhf(xe[5] * scale[5]);
  float v6 = tanhf(xe[6] * scale[6]);
  float v7 = tanhf(xe[7] * scale[7]);
  float v8 = tanhf(xe[8] * scale[8]);
  float v9 = tanhf(xe[9] * scale[9]);
  float s = v0*v0 + v1*v1 + v2*v2 + v3*v3 + v4*v4
          + v5*v5 + v6*v6 + v7*v7 + v8*v8 + v9*v9;
  float n1sq = s + 1e-8f;          // norm1^2
  float s2   = s / n1sq;
  float n2sq = s2 + 1e-8f;         // norm2^2
  float s3   = s2 / n2sq;          // norm3^2
  float inv  = 1.0f / sqrtf(n1sq * n2sq * s3);
  v0 *= inv; v1 *= inv; v2 *= inv; v3 *= inv; v4 *= inv;
  v5 *= inv; v6 *= inv; v7 *= inv; v8 *= inv; v9 *= inv;

  // ---- y = ReA * v  (D = A(16x16) x B(16x16 of per-element v-columns))
  // A-operand 16x4 chunk t: lane m holds {A[m][4t+0],A[m][4t+1]},
  //                         lane m+16 holds {A[m][4t+2],A[m][4t+3]}.
  // B-operand 4x16 chunk t: lane n holds {v_n[4t+0],v_n[4t+1]},
  //                         lane n+16 holds {v_n[4t+2],v_n[4t+3]}.
  const float* Arow = ReA + col * 16 + (hi ? 2 : 0);
  v2f a0, a1, a2, b0, b1, b2;
  a0.x = Arow[0]; a0.y = Arow[1];
  a1.x = Arow[4]; a1.y = Arow[5];
  a2.x = Arow[8]; a2.y = Arow[9];
  b0.x = hi ? v2 : v0;   b0.y = hi ? v3 : v1;
  b1.x = hi ? v6 : v4;   b1.y = hi ? v7 : v5;
  b2.x = hi ? 0.f : v8;  b2.y = hi ? 0.f : v9;   // K=10,11 components are 0

  v8f acc = {};
  acc = __builtin_amdgcn_wmma_f32_16x16x4_f32(false, a0, false, b0, (short)0, acc, false, false);
  acc = __builtin_amdgcn_wmma_f32_16x16x4_f32(false, a1, false, b1, (short)0, acc, false, false);
  acc = __builtin_amdgcn_wmma_f32_16x16x4_f32(false, a2, false, b2, (short)0, acc, false, false);
  // K chunk 12..15: B operand identically zero -> WMMA elided.

  // ---- z = v . y : lanes 0-15 hold rows 0-7 of column `col`, lanes 16-31 rows 8-15
  float plo = acc[0]*v0 + acc[1]*v1 + acc[2]*v2 + acc[3]*v3
            + acc[4]*v4 + acc[5]*v5 + acc[6]*v6 + acc[7]*v7;
  float phi = acc[0]*v8 + acc[1]*v9;      // rows 10..15 carry zero A -> zero y
  float part = hi ? phi : plo;
  float z = part + __shfl_xor(part, 16, 32);

  if (!hi) {
    float w0 = out_w[0], w1 = out_w[1];
    float b0s = out_b[0], b1s = out_b[1];
    out[e * 2 + 0] = fmaf(z, w0, b0s);
    out[e * 2 + 1] = fmaf(z, w1, b1s);
  }
}

// ---------------------------------------------------------------------------
extern "C" void kernel_launch(void* const* d_in, const int* in_sizes, int n_in,
                              void* d_out, int out_size, void* d_ws, size_t ws_size,
                              hipStream_t stream) {
  (void)in_sizes; (void)n_in; (void)out_size; (void)ws_size;
  const float* x     = (const float*)d_in[0];
  const float* scale = (const float*)d_in[1];
  const float* theta = (const float*)d_in[2];
  const float* out_w = (const float*)d_in[3];
  const float* out_b = (const float*)d_in[4];
  float* out = (float*)d_out;

  // workspace: W (1024 x 10 complex64 = 80 KB) then ReA (16x16 f32 = 1 KB)
  float2* W   = (float2*)d_ws;
  float*  ReA = (float*)((char*)d_ws + (size_t)DIM * NQ * sizeof(float2));

  qc_sim_basis<<<NQ, 512, 0, stream>>>(theta, W);
  qc_build_A<<<1, 256, 0, stream>>>(W, ReA);

  const int block = 256;                          // 8 waves/block
  const int grid  = (BATCH / 16) * 32 / block;    // 2048 waves -> 256 blocks
  qc_batch<<<grid, block, 0, stream>>>(x, scale, out_w, out_b, ReA, out);
}